// ViGBlock_15942918603269
// MI455X (gfx1250) — compile-verified
//
#include <hip/hip_runtime.h>
#include <hip/hip_bf16.h>

// ---------------------------------------------------------------------------
// ViG block for gfx1250 (CDNA5). fp32 end-to-end using V_WMMA_F32_16X16X4_F32.
// ---------------------------------------------------------------------------

#define BB    8
#define CCH   100
#define NPTS  3136          // 56*56
#define KNN   9
#define HG    200           // 2*C
#define HF    400           // 4*C
#define MROWS (BB * NPTS)   // 25088
#define CHUNK 784           // NPTS/4, multiple of 16
#define CHUNK_PAD 788
#define CTILES (CHUNK / 16) // 49

typedef float v2f __attribute__((ext_vector_type(2)));
typedef float v8f __attribute__((ext_vector_type(8)));

// ---------------------------------------------------------------------------
// fp32 WMMA GEMM: Out[M x N] = A[M x K] @ W[K x N] (+ bias), W given TRANSPOSED
// (WT is N x K row-major). One wave per 16x32 output tile (two accumulators).
// M % 16 == 0, K % 4 == 0. Out-of-range N columns handled by clamped loads
// (only pollutes unstored columns) -> branch-free inner loop.
// ---------------------------------------------------------------------------
__global__ void wmma_gemm_f32(const float* __restrict__ A,
                              const float* __restrict__ WT,
                              const float* __restrict__ bias,
                              float* __restrict__ Out,
                              int Mrows, int Kdim, int Ncols, int Ntiles2) {
  int waveg = blockIdx.x * (blockDim.x >> 5) + (threadIdx.x >> 5);
  int mt = waveg / Ntiles2;
  int nt = waveg - mt * Ntiles2;
  if (mt * 16 >= Mrows) return;               // wave-uniform guard

  int lane  = threadIdx.x & 31;
  int lrow  = lane & 15;                      // M (A) / N (B) index within tile
  int khalf = lane >> 4;                      // selects K offset 0 or 2

  const float* arow = A + (size_t)(mt * 16 + lrow) * Kdim + khalf * 2;
  int c0 = nt * 32 + lrow;
  int c1 = c0 + 16;
  int cc0 = c0 < Ncols ? c0 : Ncols - 1;      // clamp: branch-free loads
  int cc1 = c1 < Ncols ? c1 : Ncols - 1;
  const float* w0p = WT + (size_t)cc0 * Kdim + khalf * 2;
  const float* w1p = WT + (size_t)cc1 * Kdim + khalf * 2;

  v8f acc0 = {}, acc1 = {};
  for (int k = 0; k < Kdim; k += 4) {
    v2f a  = *(const v2f*)(arow + k);         // A[m][k+2h .. k+2h+1]
    v2f w0 = *(const v2f*)(w0p + k);          // WT[n0][k+2h .. +1]
    v2f w1 = *(const v2f*)(w1p + k);          // WT[n1][k+2h .. +1]
    acc0 = __builtin_amdgcn_wmma_f32_16x16x4_f32(
        false, a, false, w0, (short)0, acc0, false, false);
    acc1 = __builtin_amdgcn_wmma_f32_16x16x4_f32(
        false, a, false, w1, (short)0, acc1, false, false);
  }

  int mbase = mt * 16 + 8 * khalf;            // rows r + 8*khalf
  if (c0 < Ncols) {
    float bv = bias ? bias[c0] : 0.0f;
#pragma unroll
    for (int r = 0; r < 8; ++r)
      Out[(size_t)(mbase + r) * Ncols + c0] = acc0[r] + bv;
  }
  if (c1 < Ncols) {
    float bv = bias ? bias[c1] : 0.0f;
#pragma unroll
    for (int r = 0; r < 8; ++r)
      Out[(size_t)(mbase + r) * Ncols + c1] = acc1[r] + bv;
  }
}

// W[K x N] -> WT[N x K] (row-major both); contiguous writes.
__global__ void transpose_w_kernel(const float* __restrict__ W,
                                   float* __restrict__ WT, int K, int N) {
  int i = blockIdx.x * blockDim.x + threadIdx.x;
  if (i >= K * N) return;
  int n = i / K;
  int k = i - n * K;
  WT[i] = W[k * N + n];
}

// wdT[n][k] = wg[k][n] - wg[k+C][n]  (fused diff + transpose)
__global__ void make_wdT_kernel(const float* __restrict__ wg,
                                float* __restrict__ wdT) {
  int i = blockIdx.x * blockDim.x + threadIdx.x;
  if (i >= CCH * HG) return;
  int n = i / CCH;
  int k = i - n * CCH;
  wdT[i] = wg[k * HG + n] - wg[(k + CCH) * HG + n];
}

// wbT[n][k] = wg[k+C][n]
__global__ void make_wbT_kernel(const float* __restrict__ wg,
                                float* __restrict__ wbT) {
  int i = blockIdx.x * blockDim.x + threadIdx.x;
  if (i >= CCH * HG) return;
  int n = i / CCH;
  int k = i - n * CCH;
  wbT[i] = wg[(k + CCH) * HG + n];
}

// ---------------------------------------------------------------------------
// Fused KNN: per (batch, 16-row tile) compute distances via WMMA Gram tiles
// into LDS chunks, keep running top-9 per row, write indices. 256 thr/block.
// Two column tiles per wave iteration (shared A fragment, 2 indep. acc).
// ---------------------------------------------------------------------------
__global__ void knn_topk_kernel(const float* __restrict__ t1,
                                const float* __restrict__ x2,
                                int* __restrict__ idxout) {
  extern __shared__ float smem[];
  float* dist  = smem;                        // 16 x CHUNK_PAD
  float* candd = dist + 16 * CHUNK_PAD;       // 16*16*KNN floats
  int*   candi = (int*)(candd + 16 * 16 * KNN);

  int b  = blockIdx.y;
  int mt = blockIdx.x;                        // row tile in batch (0..195)
  const float* tb  = t1 + (size_t)b * NPTS * CCH;
  const float* x2b = x2 + (size_t)b * NPTS;

  int tid   = threadIdx.x;
  int wave  = tid >> 5;
  int lane  = tid & 31;
  int lrow  = lane & 15;
  int khalf = lane >> 4;
  int row   = tid >> 4;                       // scan-phase row 0..15
  int sub   = tid & 15;                       // 16 threads per row

  const float* arow = tb + (size_t)(mt * 16 + lrow) * CCH + khalf * 2;

  float bd[KNN]; int bi[KNN];
#pragma unroll
  for (int s = 0; s < KNN; ++s) { bd[s] = 3.4e38f; bi[s] = 0; }

  for (int chunk = 0; chunk < NPTS; chunk += CHUNK) {
    // ---- distance tiles: 8 waves, 2 tiles per iteration ----
    for (int ct = wave * 2; ct < CTILES; ct += 16) {
      bool two = (ct + 1) < CTILES;           // wave-uniform
      int c0 = chunk + ct * 16;
      const float* b0 = tb + (size_t)(c0 + lrow) * CCH + khalf * 2;
      const float* b1 = tb + (size_t)(c0 + (two ? 16 : 0) + lrow) * CCH + khalf * 2;
      v8f acc0 = {}, acc1 = {};
      for (int k = 0; k < CCH; k += 4) {
        v2f a  = *(const v2f*)(arow + k);
        v2f w0 = *(const v2f*)(b0 + k);
        v2f w1 = *(const v2f*)(b1 + k);
        acc0 = __builtin_amdgcn_wmma_f32_16x16x4_f32(
            false, a, false, w0, (short)0, acc0, false, false);
        acc1 = __builtin_amdgcn_wmma_f32_16x16x4_f32(
            false, a, false, w1, (short)0, acc1, false, false);
      }
      float xc0 = x2b[c0 + lrow];
#pragma unroll
      for (int r = 0; r < 8; ++r) {
        int rr = r + 8 * khalf;
        dist[rr * CHUNK_PAD + ct * 16 + lrow] = xc0 - 2.0f * acc0[r];
      }
      if (two) {
        float xc1 = x2b[c0 + 16 + lrow];
#pragma unroll
        for (int r = 0; r < 8; ++r) {
          int rr = r + 8 * khalf;
          dist[rr * CHUNK_PAD + (ct + 1) * 16 + lrow] = xc1 - 2.0f * acc1[r];
        }
      }
    }
    __syncthreads();
    // ---- running top-9 per row ----
    for (int c = sub; c < CHUNK; c += 16) {
      float dv = dist[row * CHUNK_PAD + c];
      if (dv < bd[KNN - 1]) {
        int g = chunk + c;
        int p = KNN - 1;
        while (p > 0 && bd[p - 1] > dv) {
          bd[p] = bd[p - 1]; bi[p] = bi[p - 1]; --p;
        }
        bd[p] = dv; bi[p] = g;
      }
    }
    __syncthreads();
  }

  // ---- merge 16 partial lists per row -> 9 smallest ----
#pragma unroll
  for (int s = 0; s < KNN; ++s) {
    candd[(row * 16 + sub) * KNN + s] = bd[s];
    candi[(row * 16 + sub) * KNN + s] = bi[s];
  }
  __syncthreads();
  if (sub == 0) {
    int base = row * 16 * KNN;
    for (int s = 0; s < KNN; ++s) {
      float best = 3.4e38f; int besti = 0, bestp = 0;
      for (int q = 0; q < 16 * KNN; ++q) {
        float v = candd[base + q];
        if (v < best) { best = v; besti = candi[base + q]; bestp = q; }
      }
      candd[base + bestp] = 3.4e38f;
      idxout[((size_t)b * NPTS + mt * 16 + row) * KNN + s] = besti;
    }
  }
}

// ---------------------------------------------------------------------------
// Elementwise / reduction helpers
// ---------------------------------------------------------------------------
__global__ void nchw_to_nlc_kernel(const float* __restrict__ x,
                                   float* __restrict__ t) {
  int i = blockIdx.x * blockDim.x + threadIdx.x;
  if (i >= BB * NPTS * CCH) return;
  int c = i % CCH;
  int n = (i / CCH) % NPTS;
  int b = i / (CCH * NPTS);
  t[i] = x[((size_t)b * CCH + c) * NPTS + n];
}

__global__ void nlc_to_nchw_kernel(const float* __restrict__ t,
                                   float* __restrict__ x) {
  int i = blockIdx.x * blockDim.x + threadIdx.x;
  if (i >= BB * NPTS * CCH) return;
  int n = i % NPTS;
  int c = (i / NPTS) % CCH;
  int b = i / (NPTS * CCH);
  x[i] = t[((size_t)b * NPTS + n) * CCH + c];
}

__global__ void rowsq_kernel(const float* __restrict__ t1,
                             float* __restrict__ x2) {
  int r = blockIdx.x * blockDim.x + threadIdx.x;
  if (r >= MROWS) return;
  const float* p = t1 + (size_t)r * CCH;
  float s = 0.0f;
  for (int c = 0; c < CCH; ++c) s += p[c] * p[c];
  x2[r] = s;
}

// 4 columns per block; lanes interleave columns for coalesced reads.
__global__ void bn_stats_kernel(const float* __restrict__ in,
                                float* __restrict__ mean,
                                float* __restrict__ istd,
                                int rows, int cols) {
  __shared__ float ss[256];
  __shared__ float sq[256];
  int cg  = blockIdx.x * 4;
  int sub = threadIdx.x & 3;                  // column within group
  int rt  = threadIdx.x >> 2;                 // 64 row threads
  int c   = cg + sub;
  float s = 0.0f, q = 0.0f;
  if (c < cols) {
    for (int r = rt; r < rows; r += 64) {
      float v = in[(size_t)r * cols + c];
      s += v; q += v * v;
    }
  }
  ss[threadIdx.x] = s; sq[threadIdx.x] = q;
  __syncthreads();
  for (int off = 128; off >= 4; off >>= 1) {
    if (threadIdx.x < off) {
      ss[threadIdx.x] += ss[threadIdx.x + off];
      sq[threadIdx.x] += sq[threadIdx.x + off];
    }
    __syncthreads();
  }
  if (threadIdx.x < 4 && (cg + threadIdx.x) < cols) {
    float m = ss[threadIdx.x] / rows;
    float v = sq[threadIdx.x] / rows - m * m;
    mean[cg + threadIdx.x] = m;
    istd[cg + threadIdx.x] = rsqrtf(v + 1e-5f);
  }
}

// out[i] = gamma*(in-mean)*istd + beta  [gelu?]  [+resid?]
__global__ void bn_apply_kernel(const float* __restrict__ in,
                                const float* __restrict__ mean,
                                const float* __restrict__ istd,
                                const float* __restrict__ gamma,
                                const float* __restrict__ beta,
                                const float* __restrict__ resid,
                                float* __restrict__ out,
                                size_t total, int cols, int do_gelu) {
  size_t i = (size_t)blockIdx.x * blockDim.x + threadIdx.x;
  if (i >= total) return;
  int c = (int)(i % cols);
  float v = gamma[c] * (in[i] - mean[c]) * istd[c] + beta[c];
  if (do_gelu) v = 0.5f * v * (1.0f + erff(v * 0.70710678118f));
  if (resid) v += resid[i];
  out[i] = v;
}

// h[row][d] = Abuf[row][d] + max_k Bbuf[b, idx[row][k]][d] + bg[d]
__global__ void edge_max_kernel(const float* __restrict__ Abuf,
                                const float* __restrict__ Bbuf,
                                const int* __restrict__ idx,
                                const float* __restrict__ bg,
                                float* __restrict__ h) {
  size_t i = (size_t)blockIdx.x * blockDim.x + threadIdx.x;
  if (i >= (size_t)MROWS * HG) return;
  int d = (int)(i % HG);
  size_t rn = i / HG;
  int b = (int)(rn / NPTS);
  const int* ip = idx + rn * KNN;
  float mx = -3.4e38f;
#pragma unroll
  for (int k = 0; k < KNN; ++k) {
    int j = ip[k];
    float v = Bbuf[((size_t)b * NPTS + j) * HG + d];
    mx = fmaxf(mx, v);
  }
  h[i] = Abuf[i] + mx + bg[d];
}

// ---------------------------------------------------------------------------
// Host-side helpers
// ---------------------------------------------------------------------------
static inline void launch_gemm(const float* A, const float* WT, const float* bias,
                               float* Out, int M, int K, int N, hipStream_t s) {
  int mt = M / 16, nt2 = (N + 31) / 32;
  int waves = mt * nt2;
  int blocks = (waves + 7) / 8;
  wmma_gemm_f32<<<blocks, 256, 0, s>>>(A, WT, bias, Out, M, K, N, nt2);
}

static inline void launch_tw(const float* W, float* WT, int K, int N,
                             hipStream_t s) {
  int tot = K * N;
  transpose_w_kernel<<<(tot + 255) / 256, 256, 0, s>>>(W, WT, K, N);
}

static inline void launch_bn(float* buf, const float* gamma, const float* beta,
                             const float* resid, float* mean, float* istd,
                             int cols, int do_gelu, hipStream_t s) {
  bn_stats_kernel<<<(cols + 3) / 4, 256, 0, s>>>(buf, mean, istd, MROWS, cols);
  size_t total = (size_t)MROWS * cols;
  int blocks = (int)((total + 255) / 256);
  bn_apply_kernel<<<blocks, 256, 0, s>>>(buf, mean, istd, gamma, beta, resid,
                                         buf, total, cols, do_gelu);
}

extern "C" void kernel_launch(void* const* d_in, const int* in_sizes, int n_in,
                              void* d_out, int out_size, void* d_ws, size_t ws_size,
                              hipStream_t stream) {
  const float* x   = (const float*)d_in[0];
  const float* w1  = (const float*)d_in[1];
  const float* b1  = (const float*)d_in[2];
  const float* g1  = (const float*)d_in[3];
  const float* be1 = (const float*)d_in[4];
  const float* wg  = (const float*)d_in[5];
  const float* bg  = (const float*)d_in[6];
  const float* gg  = (const float*)d_in[7];
  const float* beg = (const float*)d_in[8];
  const float* w2  = (const float*)d_in[9];
  const float* b2  = (const float*)d_in[10];
  const float* g2  = (const float*)d_in[11];
  const float* be2 = (const float*)d_in[12];
  const float* wf1 = (const float*)d_in[13];
  const float* bf1 = (const float*)d_in[14];
  const float* gf1 = (const float*)d_in[15];
  const float* bef1= (const float*)d_in[16];
  const float* wf2 = (const float*)d_in[17];
  const float* bf2 = (const float*)d_in[18];
  const float* gf2 = (const float*)d_in[19];
  const float* bef2= (const float*)d_in[20];
  float* out = (float*)d_out;

  // ---- workspace layout (floats) ----
  float* ws = (float*)d_ws;
  size_t off = 0;
  float* t    = ws + off; off += (size_t)MROWS * CCH;   // shortcut
  float* t1   = ws + off; off += (size_t)MROWS * CCH;   // BN'd projection
  float* x2   = ws + off; off += MROWS;
  float* wt   = ws + off; off += (size_t)HF * CCH;      // transposed weight scratch (40K)
  float* Abuf = ws + off; off += (size_t)MROWS * HG;    // t1@wd, then h in place
  float* Bbuf = ws + off; off += (size_t)MROWS * HG;    // t1@wg_bot
  float* t2   = ws + off; off += (size_t)MROWS * CCH;
  float* fb   = ws + off; off += (size_t)MROWS * HF;
  float* ob   = ws + off; off += (size_t)MROWS * CCH;
  float* mean = ws + off; off += 512;
  float* istd = ws + off; off += 512;
  int*   idx  = (int*)(ws + off);                       // MROWS*KNN ints

  const int TOT_BNC = BB * NPTS * CCH;

  // 1) (B,C,H,W) -> (B,N,C)
  nchw_to_nlc_kernel<<<(TOT_BNC + 255) / 256, 256, 0, stream>>>(x, t);

  // 2) t1 = BN(t @ w1 + b1)
  launch_tw(w1, wt, CCH, CCH, stream);
  launch_gemm(t, wt, b1, t1, MROWS, CCH, CCH, stream);
  launch_bn(t1, g1, be1, nullptr, mean, istd, CCH, 0, stream);

  // 3) KNN indices (fused WMMA Gram + LDS top-9)
  rowsq_kernel<<<(MROWS + 255) / 256, 256, 0, stream>>>(t1, x2);
  {
    size_t shmem = (size_t)16 * CHUNK_PAD * sizeof(float)
                 + (size_t)16 * 16 * KNN * sizeof(float)
                 + (size_t)16 * 16 * KNN * sizeof(int);
    dim3 grid(NPTS / 16, BB);
    knn_topk_kernel<<<grid, 256, shmem, stream>>>(t1, x2, idx);
  }

  // 4) edge conv: h = max_k(edge @ wg) + bg  via algebraic split
  //    xi term uses (wg_top - wg_bot), xj term uses wg_bot; both transposed.
  make_wdT_kernel<<<(CCH * HG + 255) / 256, 256, 0, stream>>>(wg, wt);
  launch_gemm(t1, wt, nullptr, Abuf, MROWS, CCH, HG, stream);
  make_wbT_kernel<<<(CCH * HG + 255) / 256, 256, 0, stream>>>(wg, wt);
  launch_gemm(t1, wt, nullptr, Bbuf, MROWS, CCH, HG, stream);
  {
    size_t total = (size_t)MROWS * HG;
    edge_max_kernel<<<(int)((total + 255) / 256), 256, 0, stream>>>(
        Abuf, Bbuf, idx, bg, Abuf);   // h in place in Abuf
  }
  // h = gelu(BN(h))
  launch_bn(Abuf, gg, beg, nullptr, mean, istd, HG, 1, stream);

  // 5) t2 = BN(h @ w2 + b2) + shortcut
  launch_tw(w2, wt, HG, CCH, stream);
  launch_gemm(Abuf, wt, b2, t2, MROWS, HG, CCH, stream);
  launch_bn(t2, g2, be2, t, mean, istd, CCH, 0, stream);

  // 6) FFN: f = gelu(BN(t2 @ wf1 + bf1))
  launch_tw(wf1, wt, CCH, HF, stream);
  launch_gemm(t2, wt, bf1, fb, MROWS, CCH, HF, stream);
  launch_bn(fb, gf1, bef1, nullptr, mean, istd, HF, 1, stream);

  // 7) out = BN(f @ wf2 + bf2) + t2
  launch_tw(wf2, wt, HF, CCH, stream);
  launch_gemm(fb, wt, bf2, ob, MROWS, HF, CCH, stream);
  launch_bn(ob, gf2, bef2, t2, mean, istd, CCH, 0, stream);

  // 8) (B,N,C) -> (B,C,H,W)
  nlc_to_nchw_kernel<<<(TOT_BNC + 255) / 256, 256, 0, stream>>>(ob, out);
}